// BasicGNN_41248865911519
// MI455X (gfx1250) — compile-verified
//
#include <hip/hip_runtime.h>
#include <math.h>

// ---------------- problem constants ----------------
#define Nn     10000
#define Ee     160000
#define INC    178
#define HIDC   712
#define OUTC   178
#define TD     32
#define NPAD   10112   // 79 * 128
#define KIN    192     // padded IN_C  (12*16)
#define MPAD   720     // padded HID   (45*16)
#define MOUT   192     // padded OUT_C
#define LDF    720     // leading dim of all feature buffers

#define BM 128
#define BN 48
#define KC 16
#define XSTR 20        // Xs row stride: bank-conflict free AND 16B aligned

typedef __attribute__((ext_vector_type(2))) float v2f;
typedef __attribute__((ext_vector_type(8))) float v8f;
typedef __attribute__((ext_vector_type(4))) int   i4;
typedef __attribute__((ext_vector_type(2))) int   i2;

#if defined(__HIP_DEVICE_COMPILE__) && __has_builtin(__builtin_amdgcn_wmma_f32_16x16x4_f32)
#define USE_WMMA_F32 1
#else
#define USE_WMMA_F32 0
#endif

#if defined(__HIP_DEVICE_COMPILE__) && \
    __has_builtin(__builtin_amdgcn_global_load_async_to_lds_b128) && \
    __has_builtin(__builtin_amdgcn_global_load_async_to_lds_b64) && \
    __has_builtin(__builtin_amdgcn_s_wait_asynccnt)
#define USE_ASYNC 1
#define NBUF 2
#else
#define USE_ASYNC 0
#define NBUF 1
#endif

// builtin prototypes: (vNi addrspace(1)*, vNi addrspace(3)*, imm int offset, imm int cpol)
#define GAS_B128(p) ((__attribute__((address_space(1))) i4*)(p))
#define LAS_B128(p) ((__attribute__((address_space(3))) i4*)(p))
#define GAS_B64(p)  ((__attribute__((address_space(1))) i2*)(p))
#define LAS_B64(p)  ((__attribute__((address_space(3))) i2*)(p))

// ---------------- padding / setup kernels ----------------

__global__ void padx_kernel(const float* __restrict__ x, float* __restrict__ dst) {
    int i = blockIdx.x * 256 + threadIdx.x;
    int total = NPAD * KIN;
    if (i >= total) return;
    int r = i / KIN, c = i - r * KIN;
    float v = 0.f;
    if (r < Nn && c < INC) v = x[(size_t)r * INC + c];
    dst[(size_t)r * LDF + c] = v;
}

__global__ void padw_kernel(const float* __restrict__ w, float* __restrict__ dst,
                            int K, int M, int Kp, int Mp) {
    int i = blockIdx.x * 256 + threadIdx.x;
    int total = Kp * Mp;
    if (i >= total) return;
    int k = i / Mp, m = i - k * Mp;
    float v = 0.f;
    if (k < K && m < M) v = w[(size_t)k * M + m];
    dst[i] = v;
}

__global__ void deg_kernel(const int* __restrict__ dstIdx, int* __restrict__ deg) {
    int e = blockIdx.x * 256 + threadIdx.x;
    if (e < Ee) atomicAdd(&deg[dstIdx[e]], 1);
}

__global__ void dinv_kernel(const int* __restrict__ deg, float* __restrict__ dinv) {
    int v = blockIdx.x * 256 + threadIdx.x;
    if (v < Nn) dinv[v] = rsqrtf((float)deg[v] + 1.0f);
}

__global__ __launch_bounds__(1024) void scan_kernel(const int* __restrict__ deg,
                                                    int* __restrict__ row_start, int n) {
    __shared__ int sh[1024];
    int tid = threadIdx.x;
    int carry = 0;
    if (tid == 0) row_start[0] = 0;
    for (int base = 0; base < n; base += 1024) {
        int v = (base + tid < n) ? deg[base + tid] : 0;
        __syncthreads();
        sh[tid] = v;
        __syncthreads();
        for (int off = 1; off < 1024; off <<= 1) {
            int t = (tid >= off) ? sh[tid - off] : 0;
            __syncthreads();
            sh[tid] += t;
            __syncthreads();
        }
        int incl = sh[tid];
        int total = sh[1023];
        if (base + tid < n) row_start[base + tid + 1] = carry + incl;
        carry += total;
    }
}

__global__ void fill_kernel(const int* __restrict__ src, const int* __restrict__ dst,
                            const int* __restrict__ row_start, int* __restrict__ cursor,
                            const float* __restrict__ dinv,
                            int* __restrict__ csr_src, float* __restrict__ csr_coef) {
    int e = blockIdx.x * 256 + threadIdx.x;
    if (e >= Ee) return;
    int s = src[e], d = dst[e];
    int pos = row_start[d] + atomicAdd(&cursor[d], 1);
    csr_src[pos] = s;
    csr_coef[pos] = dinv[s] * dinv[d];
}

__global__ __launch_bounds__(256) void time_kernel(const int* __restrict__ tstep,
        const float* __restrict__ w1, const float* __restrict__ b1,
        const float* __restrict__ tw0, const float* __restrict__ tw1v,
        const float* __restrict__ tb0, const float* __restrict__ tb1v,
        float* __restrict__ tadd0, float* __restrict__ tadd1) {
    __shared__ float semb[TD];
    __shared__ float st1[TD];
    int tid = threadIdx.x;
    float t = (float)tstep[0];
    if (tid < TD / 2) {
        float freq = expf((float)tid * (-logf(10000.0f) / (float)(TD / 2 - 1)));
        float a = t * freq;
        semb[tid] = sinf(a);
        semb[tid + TD / 2] = cosf(a);
    }
    __syncthreads();
    if (tid < TD) {
        float s = b1[tid];
        #pragma unroll
        for (int q = 0; q < TD; ++q) s += semb[q] * w1[q * TD + tid];
        st1[tid] = fmaxf(s, 0.f);
    }
    __syncthreads();
    for (int m = tid; m < HIDC; m += 256) {
        float s0 = tb0[m], s1 = tb1v[m];
        #pragma unroll
        for (int k = 0; k < TD; ++k) {
            float tv = st1[k];
            s0 += tv * tw0[(size_t)k * HIDC + m];
            s1 += tv * tw1v[(size_t)k * HIDC + m];
        }
        tadd0[m] = s0;
        tadd1[m] = s1;
    }
}

// ---------------- WMMA GEMM: H[NPAD x Mp] = X[NPAD x Kp] * W[Kp x Mp] ----------------
// block = 128 threads (4 wave32), tile BM=128 x BN=48; each wave: 32 rows x 48 cols.
// K staged in chunks of 16 through LDS via async global->LDS (double buffered).
__global__ __launch_bounds__(128) void gemm_kernel(const float* __restrict__ X, int ldx,
        const float* __restrict__ W, int ldw, float* __restrict__ H, int ldh, int Kp) {
    __shared__ __align__(16) float Xs[NBUF][BM][XSTR];
    __shared__ __align__(16) float Ws[NBUF][KC][BN];
    const int tid = threadIdx.x;
    const int rowblk = blockIdx.y * BM;
    const int colblk = blockIdx.x * BN;
    const int NC = Kp / KC;

#if USE_WMMA_F32
    const int wv   = tid >> 5;
    const int lane = tid & 31;
    const int half = lane >> 4;
    const int idx  = lane & 15;
    v8f acc[2][3];
    #pragma unroll
    for (int rt = 0; rt < 2; ++rt)
        #pragma unroll
        for (int t = 0; t < 3; ++t)
            #pragma unroll
            for (int j = 0; j < 8; ++j) acc[rt][t][j] = 0.f;
#else
    const int fr = tid;            // 1 thread per row, 48 cols each
    float freg[BN];
    #pragma unroll
    for (int j = 0; j < BN; ++j) freg[j] = 0.f;
#endif

#if USE_ASYNC
    // uniform 7 async ops per thread (and per wave) per chunk:
    //   X tile: 128x16 f32 = 512 x b128  -> 4/thread
    //   W tile:  16x48 f32 = 384 x b64   -> 3/thread
    auto issue = [&](int k0, int b) {
        #pragma unroll
        for (int q = 0; q < 4; ++q) {
            int o = tid + 128 * q;
            int r = o >> 2, cq = o & 3;
            const float* g = X + (size_t)(rowblk + r) * ldx + (k0 + cq * 4);
            __builtin_amdgcn_global_load_async_to_lds_b128(GAS_B128(g),
                LAS_B128(&Xs[b][r][cq * 4]), 0, 0);
        }
        #pragma unroll
        for (int q = 0; q < 3; ++q) {
            int o = tid + 128 * q;
            int k = o / 24, cp = o - 24 * k;
            const float* g = W + (size_t)(k0 + k) * ldw + (colblk + cp * 2);
            __builtin_amdgcn_global_load_async_to_lds_b64(GAS_B64(g),
                LAS_B64(&Ws[b][k][cp * 2]), 0, 0);
        }
    };
#else
    auto stage = [&](int k0) {
        float4 xv[4];
        float2 wv2[3];
        #pragma unroll
        for (int q = 0; q < 4; ++q) {
            int o = tid + 128 * q;
            int r = o >> 2, cq = o & 3;
            xv[q] = *(const float4*)(X + (size_t)(rowblk + r) * ldx + (k0 + cq * 4));
        }
        #pragma unroll
        for (int q = 0; q < 3; ++q) {
            int o = tid + 128 * q;
            int k = o / 24, cp = o - 24 * k;
            wv2[q] = *(const float2*)(W + (size_t)(k0 + k) * ldw + (colblk + cp * 2));
        }
        #pragma unroll
        for (int q = 0; q < 4; ++q) {
            int o = tid + 128 * q;
            int r = o >> 2, cq = o & 3;
            *(float4*)&Xs[0][r][cq * 4] = xv[q];
        }
        #pragma unroll
        for (int q = 0; q < 3; ++q) {
            int o = tid + 128 * q;
            int k = o / 24, cp = o - 24 * k;
            *(float2*)&Ws[0][k][cp * 2] = wv2[q];
        }
    };
#endif

    auto compute = [&](int b) {
#if USE_WMMA_F32
        #pragma unroll
        for (int kk = 0; kk < KC; kk += 4) {
            // A frag (16x4 f32): row = lane&15, K = kk + 2*(lane>>4) + {0,1}
            v2f a0, a1;
            a0.x = Xs[b][wv * 32 + idx][kk + half * 2 + 0];
            a0.y = Xs[b][wv * 32 + idx][kk + half * 2 + 1];
            a1.x = Xs[b][wv * 32 + 16 + idx][kk + half * 2 + 0];
            a1.y = Xs[b][wv * 32 + 16 + idx][kk + half * 2 + 1];
            #pragma unroll
            for (int t = 0; t < 3; ++t) {
                // B frag (4x16 f32): col = lane&15, same K striping
                v2f bb;
                bb.x = Ws[b][kk + half * 2 + 0][t * 16 + idx];
                bb.y = Ws[b][kk + half * 2 + 1][t * 16 + idx];
                acc[0][t] = __builtin_amdgcn_wmma_f32_16x16x4_f32(
                    false, a0, false, bb, (short)0, acc[0][t], false, false);
                acc[1][t] = __builtin_amdgcn_wmma_f32_16x16x4_f32(
                    false, a1, false, bb, (short)0, acc[1][t], false, false);
            }
        }
#else
        #pragma unroll
        for (int kk = 0; kk < KC; ++kk) {
            float xv = Xs[b][fr][kk];
            #pragma unroll
            for (int j = 0; j < BN; ++j) freg[j] += xv * Ws[b][kk][j];
        }
#endif
    };

#if USE_ASYNC
    issue(0, 0);
    for (int c = 0; c < NC; ++c) {
        int b = c & 1;
        if (c + 1 < NC) {
            issue((c + 1) * KC, b ^ 1);
            __builtin_amdgcn_s_wait_asynccnt(7);   // previous chunk's 7 ops complete
        } else {
            __builtin_amdgcn_s_wait_asynccnt(0);
        }
        __syncthreads();
        compute(b);
        __syncthreads();   // protect buffer b before iteration c+1 overwrites it
    }
#else
    for (int c = 0; c < NC; ++c) {
        stage(c * KC);
        __syncthreads();
        compute(0);
        __syncthreads();
    }
#endif

#if USE_WMMA_F32
    #pragma unroll
    for (int rt = 0; rt < 2; ++rt)
        #pragma unroll
        for (int t = 0; t < 3; ++t)
            #pragma unroll
            for (int j = 0; j < 8; ++j) {
                int m = j + 8 * half;      // C layout: m = vgpr + 8*(lane>>4)
                H[(size_t)(rowblk + wv * 32 + rt * 16 + m) * ldh + (colblk + t * 16 + idx)]
                    = acc[rt][t][j];
            }
#else
    #pragma unroll
    for (int j = 0; j < BN; ++j)
        H[(size_t)(rowblk + fr) * ldh + (colblk + j)] = freg[j];
#endif
}

// ---------------- CSR aggregation (+bias, +relu), atomic-free ----------------
__global__ __launch_bounds__(256) void agg_kernel(const float* __restrict__ H,
        const int* __restrict__ rs, const int* __restrict__ csrc,
        const float* __restrict__ ccoef, const float* __restrict__ dinv,
        const float* __restrict__ bias, float* __restrict__ out, int ldo,
        int Mreal, int Mpad, int relu, int writepad) {
    int v = blockIdx.y;
    int m = blockIdx.x * 256 + threadIdx.x;
    if (m >= Mpad) return;
    if (v >= Nn) { out[(size_t)v * ldo + m] = 0.f; return; }        // pad rows -> 0
    if (m >= Mreal) { if (writepad) out[(size_t)v * ldo + m] = 0.f; return; }
    float dv = dinv[v];
    float s = H[(size_t)v * LDF + m] * dv * dv;                      // self loop
    int e0 = rs[v], e1 = rs[v + 1];
    for (int e = e0; e < e1; ++e)
        s += H[(size_t)csrc[e] * LDF + m] * ccoef[e];
    s += bias[m];
    if (relu) s = fmaxf(s, 0.f);
    out[(size_t)v * ldo + m] = s;
}

// ---------------- in-place LayerNorm (+optional time-embedding add) ----------------
__global__ __launch_bounds__(256) void ln_kernel(float* __restrict__ X,
        const float* __restrict__ g, const float* __restrict__ b,
        const float* __restrict__ tadd) {
    __shared__ float red[256];
    int v = blockIdx.x;
    int tid = threadIdx.x;
    float vals[3];
    float s = 0.f, s2 = 0.f;
    #pragma unroll
    for (int q = 0; q < 3; ++q) {
        int m = tid + 256 * q;
        float x = (m < HIDC) ? X[(size_t)v * LDF + m] : 0.f;
        vals[q] = x;
        s += x; s2 += x * x;
    }
    red[tid] = s; __syncthreads();
    for (int off = 128; off > 0; off >>= 1) { if (tid < off) red[tid] += red[tid + off]; __syncthreads(); }
    float mu = red[0] / (float)HIDC;
    __syncthreads();
    red[tid] = s2; __syncthreads();
    for (int off = 128; off > 0; off >>= 1) { if (tid < off) red[tid] += red[tid + off]; __syncthreads(); }
    float var = red[0] / (float)HIDC - mu * mu;
    float rstd = rsqrtf(var + 1e-5f);
    #pragma unroll
    for (int q = 0; q < 3; ++q) {
        int m = tid + 256 * q;
        if (m < HIDC) {
            float y = (vals[q] - mu) * rstd * g[m] + b[m];
            if (tadd) y += tadd[m];
            X[(size_t)v * LDF + m] = y;
        }
    }
}

// ---------------- host orchestration ----------------
extern "C" void kernel_launch(void* const* d_in, const int* in_sizes, int n_in,
                              void* d_out, int out_size, void* d_ws, size_t ws_size,
                              hipStream_t stream) {
    (void)in_sizes; (void)n_in; (void)out_size; (void)ws_size;
    const float* x     = (const float*)d_in[0];
    const int*   ei    = (const int*)d_in[1];
    const int*   srcE  = ei;
    const int*   dstE  = ei + Ee;
    const int*   tstep = (const int*)d_in[2];
    const float* tw1   = (const float*)d_in[3];
    const float* tb1   = (const float*)d_in[4];
    const float* convw[4] = {(const float*)d_in[5], (const float*)d_in[6], (const float*)d_in[7], (const float*)d_in[8]};
    const float* convb[4] = {(const float*)d_in[9], (const float*)d_in[10], (const float*)d_in[11], (const float*)d_in[12]};
    const float* c2w[2]   = {(const float*)d_in[13], (const float*)d_in[14]};
    const float* c2b[2]   = {(const float*)d_in[15], (const float*)d_in[16]};
    const float* tws[2]   = {(const float*)d_in[17], (const float*)d_in[18]};
    const float* tbs[2]   = {(const float*)d_in[19], (const float*)d_in[20]};
    const float* ln1g[2]  = {(const float*)d_in[21], (const float*)d_in[22]};
    const float* ln1b[2]  = {(const float*)d_in[23], (const float*)d_in[24]};
    const float* ln2g[2]  = {(const float*)d_in[25], (const float*)d_in[26]};
    const float* ln2b[2]  = {(const float*)d_in[27], (const float*)d_in[28]};
    float* out = (float*)d_out;

    // ---- workspace carve ----
    char* p = (char*)d_ws;
    auto carveF = [&](size_t n) { float* r = (float*)p; p += ((n * 4 + 255) & ~(size_t)255); return r; };
    auto carveI = [&](size_t n) { int*   r = (int*)p;   p += ((n * 4 + 255) & ~(size_t)255); return r; };
    const size_t FEAT = (size_t)NPAD * LDF;
    float* BUF0 = carveF(FEAT);
    float* BUF1 = carveF(FEAT);     // GEMM output, always ld = LDF
    float* BUF2 = carveF(FEAT);
    float* W0p  = carveF((size_t)KIN * MPAD);
    float* W1p  = carveF((size_t)MPAD * MPAD);
    float* W2p  = carveF((size_t)MPAD * MPAD);
    float* W3p  = carveF((size_t)MPAD * MOUT);
    float* CW0p = carveF((size_t)MPAD * MPAD);
    float* CW1p = carveF((size_t)MPAD * MPAD);
    float* tadd0 = carveF(MPAD);
    float* tadd1 = carveF(MPAD);
    float* dinv  = carveF(Nn);
    float* ccoef = carveF(Ee);
    int* deg    = carveI(Nn);
    int* rstart = carveI(Nn + 1);
    int* cursor = carveI(Nn);
    int* csrc   = carveI(Ee);

    // ---- setup: pads, degrees, CSR, time embedding ----
    padx_kernel<<<(NPAD * KIN + 255) / 256, 256, 0, stream>>>(x, BUF0);
    padw_kernel<<<((size_t)KIN * MPAD + 255) / 256, 256, 0, stream>>>(convw[0], W0p, INC,  HIDC, KIN,  MPAD);
    padw_kernel<<<((size_t)MPAD * MPAD + 255) / 256, 256, 0, stream>>>(convw[1], W1p, HIDC, HIDC, MPAD, MPAD);
    padw_kernel<<<((size_t)MPAD * MPAD + 255) / 256, 256, 0, stream>>>(convw[2], W2p, HIDC, HIDC, MPAD, MPAD);
    padw_kernel<<<((size_t)MPAD * MOUT + 255) / 256, 256, 0, stream>>>(convw[3], W3p, HIDC, OUTC, MPAD, MOUT);
    padw_kernel<<<((size_t)MPAD * MPAD + 255) / 256, 256, 0, stream>>>(c2w[0], CW0p, HIDC, HIDC, MPAD, MPAD);
    padw_kernel<<<((size_t)MPAD * MPAD + 255) / 256, 256, 0, stream>>>(c2w[1], CW1p, HIDC, HIDC, MPAD, MPAD);
    (void)hipMemsetAsync(deg, 0, Nn * sizeof(int), stream);
    (void)hipMemsetAsync(cursor, 0, Nn * sizeof(int), stream);
    deg_kernel<<<(Ee + 255) / 256, 256, 0, stream>>>(dstE, deg);
    dinv_kernel<<<(Nn + 255) / 256, 256, 0, stream>>>(deg, dinv);
    scan_kernel<<<1, 1024, 0, stream>>>(deg, rstart, Nn);
    fill_kernel<<<(Ee + 255) / 256, 256, 0, stream>>>(srcE, dstE, rstart, cursor, dinv, csrc, ccoef);
    time_kernel<<<1, 256, 0, stream>>>(tstep, tw1, tb1, tws[0], tws[1], tbs[0], tbs[1], tadd0, tadd1);

    auto gemm = [&](const float* X_, const float* W_, int Kp, int Mp) {
        dim3 g(Mp / BN, NPAD / BM);
        gemm_kernel<<<g, 128, 0, stream>>>(X_, LDF, W_, Mp, BUF1, LDF, Kp);
    };
    auto agg = [&](const float* bias, float* dst, int relu) {
        dim3 g((MPAD + 255) / 256, NPAD);
        agg_kernel<<<g, 256, 0, stream>>>(BUF1, rstart, csrc, ccoef, dinv, bias,
                                          dst, LDF, HIDC, MPAD, relu, 1);
    };

    // layer 0: conv0 + relu
    gemm(BUF0, W0p, KIN, MPAD);
    agg(convb[0], BUF2, 1);
    // layer 1: conv1 + relu ; ln1_0 + tadd0 ; conv2_0 + relu ; ln2_0
    gemm(BUF2, W1p, MPAD, MPAD);
    agg(convb[1], BUF0, 1);
    ln_kernel<<<Nn, 256, 0, stream>>>(BUF0, ln1g[0], ln1b[0], tadd0);
    gemm(BUF0, CW0p, MPAD, MPAD);
    agg(c2b[0], BUF2, 1);
    ln_kernel<<<Nn, 256, 0, stream>>>(BUF2, ln2g[0], ln2b[0], nullptr);
    // layer 2: conv2 + relu ; ln1_1 + tadd1 ; conv2_1 + relu ; ln2_1
    gemm(BUF2, W2p, MPAD, MPAD);
    agg(convb[2], BUF0, 1);
    ln_kernel<<<Nn, 256, 0, stream>>>(BUF0, ln1g[1], ln1b[1], tadd1);
    gemm(BUF0, CW1p, MPAD, MPAD);
    agg(c2b[1], BUF2, 1);
    ln_kernel<<<Nn, 256, 0, stream>>>(BUF2, ln2g[1], ln2b[1], nullptr);
    // layer 3: conv3, no relu, write straight to d_out [Nn x 178]
    gemm(BUF2, W3p, MPAD, MOUT);
    {
        dim3 g(1, Nn);
        agg_kernel<<<g, 256, 0, stream>>>(BUF1, rstart, csrc, ccoef, dinv, convb[3],
                                          out, OUTC, OUTC, OUTC, 0, 0);
    }
}